// BaseSparseConn_48576080118043
// MI455X (gfx1250) — compile-verified
//
#include <hip/hip_runtime.h>

typedef __attribute__((ext_vector_type(2))) float v2f;
typedef __attribute__((ext_vector_type(8))) float v8f;

#define BATCH 64
#define CHUNKS 4

// ---------------------------------------------------------------------------
// Kernel 1: transpose x (BATCH x NS) -> xt (NS x BATCH), LDS-tiled 32x32.
// Makes per-edge gathers contiguous 256B bursts in the scatter kernel.
// ---------------------------------------------------------------------------
__global__ void k_transpose(const float* __restrict__ x, float* __restrict__ xt, int ns) {
    __shared__ float tile[32][33];
    const int tx = threadIdx.x;   // 0..31
    const int ty = threadIdx.y;   // 0..7
    const int s0 = blockIdx.x * 32;
    const int b0 = blockIdx.y * 32;
#pragma unroll
    for (int k = 0; k < 4; ++k) {
        int s = s0 + tx, b = b0 + ty + k * 8;
        if (s < ns && b < BATCH) tile[ty + k * 8][tx] = x[(size_t)b * ns + s];
    }
    __syncthreads();
#pragma unroll
    for (int k = 0; k < 4; ++k) {
        int s = s0 + ty + k * 8, b = b0 + tx;
        if (s < ns && b < BATCH) xt[(size_t)s * BATCH + b] = tile[tx][ty + k * 8];
    }
}

// ---------------------------------------------------------------------------
// Kernel 2: zero the (ND x BATCH) accumulator (float4 stores).
// ---------------------------------------------------------------------------
__global__ void k_zero(float4* __restrict__ p, long long n4) {
    long long i = (long long)blockIdx.x * blockDim.x + threadIdx.x;
    const long long stride = (long long)gridDim.x * blockDim.x;
    const float4 z = make_float4(0.f, 0.f, 0.f, 0.f);
    for (; i < n4; i += stride) p[i] = z;
}

// ---------------------------------------------------------------------------
// Kernel 3: COO scatter. One wave per 32-edge group (CHUNKS groups/wave).
// Edge data loaded coalesced, broadcast with wave32 shuffles; each edge is a
// contiguous 256B gather from xt and 2 f32 atomic adds per lane into out_t.
// out_t + xt both fit in the 192MB L2 -> atomics & gathers resolve at L2.
// ---------------------------------------------------------------------------
__global__ void k_scatter(const float* __restrict__ xt, const float* __restrict__ vals,
                          const int* __restrict__ row, const int* __restrict__ col,
                          float* __restrict__ out_t, int nnz) {
    const int lane = threadIdx.x & 31;
    const long long wave = ((long long)blockIdx.x * blockDim.x + threadIdx.x) >> 5;
    const long long base0 = wave * (32LL * CHUNKS);
    for (int ch = 0; ch < CHUNKS; ++ch) {
        const long long base = base0 + 32LL * ch;
        if (base >= nnz) return;
        const long long idx = base + lane;
        float v = 0.f;
        int r = 0, c = 0;
        if (idx < nnz) {
            v = vals[idx];
            r = row[idx];
            c = col[idx];
        }
        if (ch + 1 < CHUNKS && base + 32 + lane < nnz) {
            // gfx1250 global_prefetch_b8: stream next edge chunk toward L2/L0
            __builtin_prefetch(vals + base + 32 + lane, 0, 0);
            __builtin_prefetch(row + base + 32 + lane, 0, 0);
            __builtin_prefetch(col + base + 32 + lane, 0, 0);
        }
        const int count = (nnz - base < 32) ? (int)(nnz - base) : 32;
        for (int j = 0; j < count; ++j) {
            const float vj = __shfl(v, j);
            const int rj   = __shfl(r, j);
            const int cj   = __shfl(c, j);
            const v2f xv = *(const v2f*)(xt + (size_t)cj * BATCH + lane * 2);
            float* dst = out_t + (size_t)rj * BATCH + lane * 2;
            unsafeAtomicAdd(dst, xv.x * vj);        // global_atomic_add_f32
            unsafeAtomicAdd(dst + 1, xv.y * vj);
        }
    }
}

// ---------------------------------------------------------------------------
// Kernel 4: WMMA epilogue. out(b,d) = out_t(d,b) + bias(d) done as
//   D = A x B + C   with  A = e0 (16x4, ones in column k=0),
//                         B row0 = bias[d0..d0+15]   (4x16),
//                         C = transposed 16x16 accumulator tile.
// One wave per 16(b) x 16(d) tile; EXEC all-1s (guards are wave-uniform and
// all tiles full: BATCH%16==0, handled d-range is a multiple of 16).
// ---------------------------------------------------------------------------
__global__ void k_epilogue_wmma(const float* __restrict__ out_t, const float* __restrict__ bias,
                                float* __restrict__ out, int nd, int dtiles) {
    const int lane = threadIdx.x & 31;
    const int wave = (int)(((long long)blockIdx.x * blockDim.x + threadIdx.x) >> 5);
    const int tiles = (BATCH / 16) * dtiles;
    if (wave >= tiles) return;  // wave-uniform exit
    const int b0 = (wave & 3) * 16;
    const int d0 = (wave >> 2) * 16;
    const int n = lane & 15;                 // column (d offset) in tile
    const int mbase = (lane < 16) ? 0 : 8;   // row block (b offset) per C/D layout

    // C-layout load: lane holds rows mbase..mbase+7 of column n -> 8 consecutive
    // floats of out_t row (d0+n), 32B aligned.
    const float* cbase = out_t + (size_t)(d0 + n) * BATCH + b0 + mbase;
    const float4 lo = *(const float4*)(cbase);
    const float4 hi = *(const float4*)(cbase + 4);
    v8f c = {lo.x, lo.y, lo.z, lo.w, hi.x, hi.y, hi.z, hi.w};

    // A: 16x4, A[m,0]=1 else 0  (K0 lives in VGPR0 of lanes 0..15)
    v2f a = {(lane < 16) ? 1.0f : 0.0f, 0.0f};
    // B: 4x16, B[0,n]=bias[d0+n] else 0 (K0 row lives in VGPR0 of lanes 0..15)
    v2f bm = {(lane < 16) ? bias[d0 + n] : 0.0f, 0.0f};

    v8f d = __builtin_amdgcn_wmma_f32_16x16x4_f32(
        /*neg_a=*/false, a, /*neg_b=*/false, bm,
        /*c_mod=*/(short)0, c, /*reuse_a=*/false, /*reuse_b=*/false);

#pragma unroll
    for (int v = 0; v < 8; ++v) {
        out[(size_t)(b0 + mbase + v) * nd + d0 + n] = d[v];  // coalesced along n
    }
}

// Scalar tail for nd % 16 != 0 (not taken for the reference shapes).
__global__ void k_epilogue_tail(const float* __restrict__ out_t, const float* __restrict__ bias,
                                float* __restrict__ out, int nd, int dstart) {
    const int total = (nd - dstart) * BATCH;
    const int i = blockIdx.x * blockDim.x + threadIdx.x;
    if (i >= total) return;
    const int d = dstart + (i >> 6);
    const int b = i & 63;
    out[(size_t)b * nd + d] = out_t[(size_t)d * BATCH + b] + bias[d];
}

// ---------------------------------------------------------------------------
// Inputs (setup_inputs order): x(B*NS f32), vals(NNZ f32), bias(ND f32),
// row(NNZ i32), col(NNZ i32). Output: (B, ND) f32.
// Workspace: xt (NS*64 f32) + out_t (ND*64 f32)  ~= 51.2 MB.
// ---------------------------------------------------------------------------
extern "C" void kernel_launch(void* const* d_in, const int* in_sizes, int n_in,
                              void* d_out, int out_size, void* d_ws, size_t ws_size,
                              hipStream_t stream) {
    const float* x    = (const float*)d_in[0];
    const float* vals = (const float*)d_in[1];
    const float* bias = (const float*)d_in[2];
    const int* row    = (const int*)d_in[3];
    const int* col    = (const int*)d_in[4];
    float* out = (float*)d_out;

    const int nnz = in_sizes[1];
    const int nd  = in_sizes[2];
    const int ns  = in_sizes[0] / BATCH;

    float* xt    = (float*)d_ws;
    float* out_t = xt + (size_t)ns * BATCH;

    // 1) transpose x -> xt
    dim3 tb(32, 8);
    dim3 tg((ns + 31) / 32, (BATCH + 31) / 32);
    k_transpose<<<tg, tb, 0, stream>>>(x, xt, ns);

    // 2) zero accumulator
    long long n4 = ((long long)nd * BATCH) / 4;
    long long zb = (n4 + 255) / 256;
    if (zb > 16384) zb = 16384;
    k_zero<<<(int)zb, 256, 0, stream>>>((float4*)out_t, n4);

    // 3) scatter: 8 waves/block, CHUNKS*32 edges per wave
    long long waves = ((long long)nnz + 32LL * CHUNKS - 1) / (32LL * CHUNKS);
    long long sblocks = (waves + 7) / 8;
    if (sblocks < 1) sblocks = 1;
    k_scatter<<<(int)sblocks, 256, 0, stream>>>(xt, vals, row, col, out_t, nnz);

    // 4) epilogue: WMMA bias-add + transpose to (B, ND)
    const int dtiles = nd / 16;
    if (dtiles > 0) {
        const int tiles = (BATCH / 16) * dtiles;
        const int eblocks = (tiles + 7) / 8;  // 8 waves (tiles) per 256-thread block
        k_epilogue_wmma<<<eblocks, 256, 0, stream>>>(out_t, bias, out, nd, dtiles);
    }
    const int dstart = dtiles * 16;
    if (dstart < nd) {
        const int total = (nd - dstart) * BATCH;
        k_epilogue_tail<<<(total + 255) / 256, 256, 0, stream>>>(out_t, bias, out, nd, dstart);
    }
}